// TiltedStableDrivenSDE_75737453298059
// MI455X (gfx1250) — compile-verified
//
#include <hip/hip_runtime.h>
#include <stdint.h>

// ---- problem constants (match reference) ----
#define STATE_DIM 256
#define T_STEPS   2048
#define MAX_JUMPS 32
#define TAU_F     0.1f
#define NEG_INV_ALPHA (-0.66666666666666666f)   // -1/1.5

// ---- scan parallelization ----
#define NCHUNK    32
#define CHUNK     (T_STEPS / NCHUNK)            // 64
#define LOOKAHEAD 8                             // outstanding async rows
#define RING      16                            // LDS ring slots (> LOOKAHEAD so the
                                                // slot written this step is never the
                                                // slot read this step: no DS/ASYNC
                                                // cross-counter ordering hazard)

static_assert(NCHUNK * CHUNK == T_STEPS, "chunking must tile T");
static_assert(RING > LOOKAHEAD, "ring must exceed pipeline depth");
static_assert((CHUNK % RING) == 0, "slot index t&(RING-1) must align per chunk");

// CDNA5 async global->LDS path (probe-confirmed: builtins declared, pointee
// type int, global src first, LDS dst second).
#if defined(__HIP_DEVICE_COMPILE__) && defined(__gfx1250__) && \
    __has_builtin(__builtin_amdgcn_global_load_async_to_lds_b32) && \
    __has_builtin(__builtin_amdgcn_s_wait_asynccnt)
#define USE_ASYNC 1
#else
#define USE_ASYNC 0
#endif

#if USE_ASYNC
__device__ __forceinline__ void async_copy_f32(const float* gsrc, const float* ldst) {
  // generic LDS pointer low 32 bits == LDS byte offset (addrspacecast layout)
  __builtin_amdgcn_global_load_async_to_lds_b32(
      (__attribute__((address_space(1))) int*)(uintptr_t)gsrc,
      (__attribute__((address_space(3))) int*)(uint32_t)(uintptr_t)ldst,
      0, 0);
}
#define WAIT_ASYNC(n) __builtin_amdgcn_s_wait_asynccnt(n)
#endif

// pow(x, -1/alpha) = exp2(-1/alpha * log2(x)) on the native transcendental unit
__device__ __forceinline__ float pow_neg_inv_alpha(float x) {
#if defined(__HIP_DEVICE_COMPILE__)
  return __builtin_amdgcn_exp2f(NEG_INV_ALPHA * __builtin_amdgcn_logf(x));
#else
  return 0.0f;  // host-side parse only; never executed
#endif
}

// ---------------------------------------------------------------------------
// K2 (defined first so the disasm snippet shows the async-to-LDS pipeline):
// per-chunk local scan with X_in = 0.
//   L_t = L_{t-1} + (w*L_{t-1}+b)*dt_t + dw*S_t   (same op order as reference)
//   P   = prod(1 + w*dt_t)
// S rows streamed global->LDS with an async ring (16 slots, 8 outstanding);
// the prologue asyncs are issued FIRST so the DMA runs under the sdt staging
// and barrier. Each wave only touches the LDS slice its own lanes cover, so
// no barriers are needed in the loop. Writes local path in place over S.
// ---------------------------------------------------------------------------
__global__ __launch_bounds__(STATE_DIM)
void k_scan_local(const float* __restrict__ ts,
                  const float* __restrict__ w_, const float* __restrict__ b_,
                  const float* __restrict__ dw_,
                  float* __restrict__ path,
                  float* __restrict__ Ec, float* __restrict__ Pc) {
  __shared__ float sS[RING * STATE_DIM];
  __shared__ float sdt[CHUNK];
  const int c  = blockIdx.x;
  const int d  = threadIdx.x;
  const int t0 = c * CHUNK;

#if USE_ASYNC
  // kick off the first LOOKAHEAD rows immediately; they land while we stage dt
  #pragma unroll
  for (int k = 0; k < LOOKAHEAD; ++k)
    async_copy_f32(&path[(t0 + k) * STATE_DIM + d],
                   &sS[((t0 + k) & (RING - 1)) * STATE_DIM + d]);
#endif

  if (d < CHUNK) {
    const int t = t0 + d;
    sdt[d] = (t == 0) ? 0.0f : (ts[t] - ts[t - 1]);
  }
  __syncthreads();

  const float w = w_[d], b = b_[d], dw = dw_[d];
  float X = 0.0f, P = 1.0f;

  for (int i = 0; i < CHUNK; ++i) {
    const int t = t0 + i;
    float Sv;
#if USE_ASYNC
    WAIT_ASYNC(LOOKAHEAD - 1);                     // oldest outstanding row landed
    Sv = sS[(t & (RING - 1)) * STATE_DIM + d];
    int tn  = t + LOOKAHEAD;
    int tcl = (tn < T_STEPS) ? tn : (T_STEPS - 1); // keep ring full (dummy ok)
    async_copy_f32(&path[tcl * STATE_DIM + d],
                   &sS[(tn & (RING - 1)) * STATE_DIM + d]);
#else
    Sv = path[t * STATE_DIM + d];
    {
      int tp = t + LOOKAHEAD;
      if (tp < T_STEPS) __builtin_prefetch(&path[tp * STATE_DIM + d], 0, 3);
    }
#endif
    const float dt    = sdt[i];
    const float a     = fmaf(w, dt, 1.0f);
    const float drift = fmaf(w, X, b);
    X = fmaf(drift, dt, X);
    X = fmaf(dw, Sv, X);
    P *= a;
    path[t * STATE_DIM + d] = X;
  }
#if USE_ASYNC
  WAIT_ASYNC(0);                                   // drain before LDS goes away
#endif
  Ec[c * STATE_DIM + d] = X;
  Pc[c * STATE_DIM + d] = P;
}

// ---------------------------------------------------------------------------
// K1: S[t,d] = TAU * sum_{j<nj} (1-u)^(-1/alpha) * g    (written into d_out)
// Memory-bound but sparse: expected 0.42 jumps per (t,d), so only read rows
// j < nj. Prefetch the j=0 rows (34% of lanes use them) to hide latency.
// ---------------------------------------------------------------------------
__global__ __launch_bounds__(STATE_DIM)
void k_jump_sums(const float* __restrict__ u, const float* __restrict__ g,
                 const int* __restrict__ nj, float* __restrict__ S) {
  const int t = blockIdx.x;
  const int d = threadIdx.x;
  const float* up = u + (size_t)t * MAX_JUMPS * STATE_DIM + d;
  const float* gp = g + (size_t)t * MAX_JUMPS * STATE_DIM + d;
  __builtin_prefetch(up, 0, 3);   // -> global_prefetch_b8
  __builtin_prefetch(gp, 0, 3);
  int n = nj[t * STATE_DIM + d];
  n = (n > MAX_JUMPS) ? MAX_JUMPS : n;
  float s = 0.0f;
  for (int j = 0; j < n; ++j) {
    const float uu = up[j * STATE_DIM];
    const float gg = gp[j * STATE_DIM];
    const float p  = pow_neg_inv_alpha(1.0f - uu);   // v_log_f32 + v_exp_f32
    s = fmaf(p, gg, s);
  }
  S[t * STATE_DIM + d] = TAU_F * s;
}

// ---------------------------------------------------------------------------
// K3: chunk-boundary prefix: Xin[0]=state_init; Xin[c]=E[c-1]+Xin[c-1]*P[c-1].
// All E/P loaded up-front (unrolled -> all loads in flight at once).
// ---------------------------------------------------------------------------
__global__ __launch_bounds__(STATE_DIM)
void k_prefix(const float* __restrict__ x0, const float* __restrict__ Ec,
              const float* __restrict__ Pc, float* __restrict__ Xin) {
  const int d = threadIdx.x;
  float e[NCHUNK], p[NCHUNK];
  #pragma unroll
  for (int c = 0; c < NCHUNK; ++c) {
    e[c] = Ec[c * STATE_DIM + d];
    p[c] = Pc[c * STATE_DIM + d];
  }
  float xin = x0[d];
  Xin[d] = xin;
  #pragma unroll
  for (int c = 1; c < NCHUNK; ++c) {
    xin = fmaf(xin, p[c - 1], e[c - 1]);
    Xin[c * STATE_DIM + d] = xin;
  }
}

// ---------------------------------------------------------------------------
// K4: fixup: path[t] += Xin_c * prod_{t0..t}(1 + w*dt). Product chain is the
// only dependence (1 mul/step); path loads are independent and prefetched.
// ---------------------------------------------------------------------------
__global__ __launch_bounds__(STATE_DIM)
void k_fixup(const float* __restrict__ ts, const float* __restrict__ w_,
             const float* __restrict__ Xin, float* __restrict__ path) {
  __shared__ float sdt[CHUNK];
  const int c  = blockIdx.x;
  const int d  = threadIdx.x;
  const int t0 = c * CHUNK;
  if (d < CHUNK) {
    const int t = t0 + d;
    sdt[d] = (t == 0) ? 0.0f : (ts[t] - ts[t - 1]);
  }
  __syncthreads();
  const float w   = w_[d];
  const float xin = Xin[c * STATE_DIM + d];
  float P = 1.0f;
  for (int i = 0; i < CHUNK; ++i) {
    const int t = t0 + i;
    {
      int tp = t + 8;
      if (tp < T_STEPS) __builtin_prefetch(&path[tp * STATE_DIM + d], 0, 3);
    }
    P *= fmaf(w, sdt[i], 1.0f);
    path[t * STATE_DIM + d] = fmaf(xin, P, path[t * STATE_DIM + d]);
  }
}

// ---------------------------------------------------------------------------
extern "C" void kernel_launch(void* const* d_in, const int* in_sizes, int n_in,
                              void* d_out, int out_size, void* d_ws, size_t ws_size,
                              hipStream_t stream) {
  (void)in_sizes; (void)n_in; (void)out_size; (void)ws_size;
  const float* state_init = (const float*)d_in[0];
  const float* ts         = (const float*)d_in[1];
  const float* w          = (const float*)d_in[2];
  const float* b          = (const float*)d_in[3];
  const float* dw         = (const float*)d_in[4];
  const float* u          = (const float*)d_in[5];
  const float* g          = (const float*)d_in[6];
  const int*   nj         = (const int*)d_in[7];

  float* path = (float*)d_out;                 // (T, D); holds S between K1/K2
  float* Ec   = (float*)d_ws;                  // NCHUNK*D
  float* Pc   = Ec + NCHUNK * STATE_DIM;       // NCHUNK*D
  float* Xin  = Pc + NCHUNK * STATE_DIM;       // NCHUNK*D   (total 96 KB)

  k_jump_sums <<<T_STEPS, STATE_DIM, 0, stream>>>(u, g, nj, path);
  k_scan_local<<<NCHUNK,  STATE_DIM, 0, stream>>>(ts, w, b, dw, path, Ec, Pc);
  k_prefix    <<<1,       STATE_DIM, 0, stream>>>(state_init, Ec, Pc, Xin);
  k_fixup     <<<NCHUNK,  STATE_DIM, 0, stream>>>(ts, w, Xin, path);
}